// weighted_CRDLoss_11295763988755
// MI455X (gfx1250) — compile-verified
//
#include <hip/hip_runtime.h>
#include <hip/hip_bf16.h>
#include <math.h>

// Problem constants (from reference)
#define BSZ   64
#define DDIM  128
#define KC    16384          // contrast K
#define KTOT  16385          // 1 + K
#define NDATA 500000
#define SDIM  1024
#define TDIM  2048

typedef __attribute__((ext_vector_type(2))) float v2f;
typedef __attribute__((ext_vector_type(8))) float v8f;

__device__ __forceinline__ float wave_reduce_add(float v) {
#pragma unroll
    for (int off = 16; off > 0; off >>= 1) v += __shfl_xor(v, off, 32);
    return v;
}

// ---------------------------------------------------------------------------
// Projection: out(64x128) = X(64xS) @ W(128xS)^T + bias, raw (un-normalized).
// One wave per 16x16 tile, V_WMMA_F32_16X16X4_F32, K-step 4.
// A 16x4 layout: lanes 0-15 -> M=lane, VGPR0=K0,VGPR1=K1; lanes 16-31 -> K2,K3.
// B 4x16 layout: VGPR0: lanes0-15 = row K0, lanes16-31 = row K1; VGPR1: K2/K3.
// D 16x16 f32: VGPR r: lanes0-15 -> M=r, lanes16-31 -> M=r+8; N = lane&15.
// ---------------------------------------------------------------------------
__global__ __launch_bounds__(32)
void proj_wmma_kernel(const float* __restrict__ X, const float* __restrict__ W,
                      const float* __restrict__ bias, float* __restrict__ out,
                      int S) {
    const int lane = threadIdx.x & 31;
    const int l15  = lane & 15;
    const int h    = lane >> 4;
    const int m0   = (blockIdx.x >> 3) << 4;   // 4 M-tiles (bsz=64)
    const int n0   = (blockIdx.x & 7) << 4;    // 8 N-tiles (d=128)

    const float* __restrict__ xrow = X + (size_t)(m0 + l15) * S;
    const float* __restrict__ wrow = W + (size_t)(n0 + l15) * S;

    v8f acc = {};
    for (int k0 = 0; k0 < S; k0 += 4) {
        v2f a, b;
        a.x = xrow[k0 + 2 * h];
        a.y = xrow[k0 + 2 * h + 1];
        b.x = wrow[k0 + h];
        b.y = wrow[k0 + 2 + h];
        acc = __builtin_amdgcn_wmma_f32_16x16x4_f32(
            /*neg_a=*/false, a, /*neg_b=*/false, b,
            /*c_mod=*/(short)0, acc, /*reuse_a=*/false, /*reuse_b=*/false);
    }
    const float bv = bias[n0 + l15];
#pragma unroll
    for (int r = 0; r < 8; ++r) {
        const int m = m0 + r + 8 * h;
        out[m * DDIM + n0 + l15] = acc[r] + bv;
    }
}

// In-place row l2norm of 128 rows (es then et), one wave per 128-float row.
__global__ __launch_bounds__(32)
void l2norm_rows_kernel(float* __restrict__ buf) {
    const int row = blockIdx.x;
    float4* p = (float4*)(buf + (size_t)row * DDIM);
    float4 v = p[threadIdx.x];
    float ss = wave_reduce_add(v.x * v.x + v.y * v.y + v.z * v.z + v.w * v.w);
    const float inv = 1.0f / sqrtf(ss);
    v.x *= inv; v.y *= inv; v.z *= inv; v.w *= inv;
    p[threadIdx.x] = v;
}

// ---------------------------------------------------------------------------
// Gathered exp-dot: outraw[b,k] = exp(dot(memA[row(b,k)], e[b]) / 0.07).
// Wave-per-k: 32 lanes read one 512B row coalesced (float4/lane), butterfly
// reduce, lane0 stores + accumulates Z partial (f64 atomic per wave).
// ---------------------------------------------------------------------------
__global__ __launch_bounds__(256)
void einsum_kernel(const float* __restrict__ memA, const float* __restrict__ evecs,
                   const int* __restrict__ idx, const int* __restrict__ cidx,
                   float* __restrict__ outraw, double* __restrict__ zsum) {
    const int b    = blockIdx.x;
    const int lane = threadIdx.x & 31;
    const int wid  = blockIdx.y * (blockDim.x >> 5) + (threadIdx.x >> 5);
    const int nw   = gridDim.y * (blockDim.x >> 5);

    const float4 ev = ((const float4*)(evecs + (size_t)b * DDIM))[lane];
    double zacc = 0.0;
    for (int k = wid; k < KTOT; k += nw) {
        const int row = (k == 0) ? idx[b] : cidx[(size_t)b * KC + (k - 1)];
        const float4 m = ((const float4*)(memA + (size_t)row * DDIM))[lane];
        float p = m.x * ev.x + m.y * ev.y + m.z * ev.z + m.w * ev.w;
        p = wave_reduce_add(p);
        const float v = expf(p / 0.07f);
        if (lane == 0) {
            outraw[(size_t)b * KTOT + k] = v;
            zacc += (double)v;
        }
    }
    if (lane == 0) atomicAdd(zsum, zacc);
}

// NCE loss partials: accumulate sum_b w_b * (log_D1 + sum_k log_D0) in f64.
__global__ __launch_bounds__(256)
void loss_kernel(const float* __restrict__ raw, const float* __restrict__ loss_s,
                 const float* __restrict__ loss_t, const double* __restrict__ zsum,
                 double* __restrict__ lacc) {
    const int which = blockIdx.z;  // 0: out_v1 with s_w, 1: out_v2 with t_w
    const int b     = blockIdx.x;
    const float w = (which == 0) ? (loss_s[b] > loss_t[b] ? 1.f : 0.f)
                                 : (loss_t[b] > loss_s[b] ? 1.f : 0.f);
    if (w == 0.0f) return;

    const float Z   = (float)(zsum[which] / ((double)BSZ * KTOT) * (double)NDATA);
    const float mPn = (float)KC / (float)NDATA;
    const float* __restrict__ x = raw + (size_t)which * (BSZ * (size_t)KTOT)
                                      + (size_t)b * KTOT;
    double acc = 0.0;
    for (int k = blockIdx.y * blockDim.x + threadIdx.x; k < KTOT;
         k += gridDim.y * blockDim.x) {
        const float xv = x[k] / Z;
        const float t = (k == 0) ? logf(xv / (xv + mPn + 1e-7f))
                                 : logf(mPn / (xv + mPn + 1e-7f));
        acc += (double)t;
    }
#pragma unroll
    for (int off = 16; off > 0; off >>= 1) acc += __shfl_xor(acc, off, 32);
    if ((threadIdx.x & 31) == 0) atomicAdd(lacc, acc);
}

// Deterministic scatter of 64 updated rows (serial over b => last dup wins).
__global__ __launch_bounds__(32)
void scatter_update_kernel(const float* __restrict__ mem1, const float* __restrict__ mem2,
                           const float* __restrict__ es, const float* __restrict__ et,
                           const int* __restrict__ idx,
                           float* __restrict__ new1, float* __restrict__ new2) {
    const int lane = threadIdx.x;
    for (int b = 0; b < BSZ; ++b) {
        const size_t row = (size_t)idx[b];
        {
            const float4 m = ((const float4*)(mem1 + row * DDIM))[lane];
            const float4 e = ((const float4*)(es + (size_t)b * DDIM))[lane];
            float4 u;
            u.x = 0.5f * m.x + 0.5f * e.x;  u.y = 0.5f * m.y + 0.5f * e.y;
            u.z = 0.5f * m.z + 0.5f * e.z;  u.w = 0.5f * m.w + 0.5f * e.w;
            const float ss = wave_reduce_add(u.x*u.x + u.y*u.y + u.z*u.z + u.w*u.w);
            const float inv = 1.0f / sqrtf(ss);
            float* d = new1 + row * DDIM + lane * 4;   // d_out+1 is only 4B aligned
            d[0] = u.x*inv; d[1] = u.y*inv; d[2] = u.z*inv; d[3] = u.w*inv;
        }
        {
            const float4 m = ((const float4*)(mem2 + row * DDIM))[lane];
            const float4 e = ((const float4*)(et + (size_t)b * DDIM))[lane];
            float4 u;
            u.x = 0.5f * m.x + 0.5f * e.x;  u.y = 0.5f * m.y + 0.5f * e.y;
            u.z = 0.5f * m.z + 0.5f * e.z;  u.w = 0.5f * m.w + 0.5f * e.w;
            const float ss = wave_reduce_add(u.x*u.x + u.y*u.y + u.z*u.z + u.w*u.w);
            const float inv = 1.0f / sqrtf(ss);
            float* d = new2 + row * DDIM + lane * 4;
            d[0] = u.x*inv; d[1] = u.y*inv; d[2] = u.z*inv; d[3] = u.w*inv;
        }
    }
}

__global__ void zero_accum_kernel(double* p) {
    if (threadIdx.x < 3) p[threadIdx.x] = 0.0;
}

__global__ void finalize_kernel(const double* __restrict__ lacc, float* __restrict__ out0) {
    out0[0] = (float)(-lacc[0] / (double)BSZ);
}

extern "C" void kernel_launch(void* const* d_in, const int* in_sizes, int n_in,
                              void* d_out, int out_size, void* d_ws, size_t ws_size,
                              hipStream_t stream) {
    const float* f_s    = (const float*)d_in[0];
    const float* f_t    = (const float*)d_in[1];
    const float* loss_s = (const float*)d_in[2];
    const float* loss_t = (const float*)d_in[3];
    const int*   idx    = (const int*)d_in[4];
    const int*   cidx   = (const int*)d_in[5];
    const float* W_s    = (const float*)d_in[6];
    const float* b_s    = (const float*)d_in[7];
    const float* W_t    = (const float*)d_in[8];
    const float* b_t    = (const float*)d_in[9];
    const float* mem1   = (const float*)d_in[10];
    const float* mem2   = (const float*)d_in[11];

    float* out  = (float*)d_out;
    float* new1 = out + 1;
    float* new2 = out + 1 + (size_t)NDATA * DDIM;

    // Workspace layout
    float*  es   = (float*)d_ws;                       // 64*128
    float*  et   = es + BSZ * DDIM;                    // 64*128
    float*  raw  = et + BSZ * DDIM;                    // 2 * 64*16385 (v1 then v2)
    double* accs = (double*)(raw + 2 * (size_t)BSZ * KTOT); // [Z1, Z2, lossAcc]

    // 1) Base copies of the memories into the outputs (dominant traffic, d2d).
    hipMemcpyAsync(new1, mem1, (size_t)NDATA * DDIM * sizeof(float),
                   hipMemcpyDeviceToDevice, stream);
    hipMemcpyAsync(new2, mem2, (size_t)NDATA * DDIM * sizeof(float),
                   hipMemcpyDeviceToDevice, stream);

    // 2) Zero f64 accumulators (Z1, Z2, loss).
    zero_accum_kernel<<<1, 32, 0, stream>>>(accs);

    // 3) Projections via f32 WMMA, then row l2norm (es and et are contiguous).
    proj_wmma_kernel<<<32, 32, 0, stream>>>(f_s, W_s, b_s, es, SDIM);
    proj_wmma_kernel<<<32, 32, 0, stream>>>(f_t, W_t, b_t, et, TDIM);
    l2norm_rows_kernel<<<2 * BSZ, 32, 0, stream>>>(es);

    // 4) Scatter the 64 momentum-updated rows (needs es/et + base copies).
    scatter_update_kernel<<<1, 32, 0, stream>>>(mem1, mem2, es, et, idx, new1, new2);

    // 5) Gathered exp-dot passes. out_v1 = f(memory_v2, es); out_v2 = f(memory_v1, et).
    einsum_kernel<<<dim3(BSZ, 8), 256, 0, stream>>>(mem2, es, idx, cidx,
                                                    raw, &accs[0]);
    einsum_kernel<<<dim3(BSZ, 8), 256, 0, stream>>>(mem1, et, idx, cidx,
                                                    raw + (size_t)BSZ * KTOT, &accs[1]);

    // 6) NCE loss reduction + finalize.
    loss_kernel<<<dim3(BSZ, 4, 2), 256, 0, stream>>>(raw, loss_s, loss_t, accs, &accs[2]);
    finalize_kernel<<<1, 1, 0, stream>>>(&accs[2], out);
}